// PointTransformerLayer_71476845740681
// MI455X (gfx1250) — compile-verified
//
#include <hip/hip_runtime.h>
#include <hip/hip_bf16.h>
#include <float.h>
#include <limits.h>

// ---------------------------------------------------------------------------
// Point Transformer layer for MI455X (gfx1250), f32 end-to-end.
// QKV GEMMs and the pass2 (256->32) projection use V_WMMA_F32_16X16X4_F32.
// ---------------------------------------------------------------------------

typedef float v2f __attribute__((ext_vector_type(2)));
typedef float v8f __attribute__((ext_vector_type(8)));

// ------------------------- WMMA f32 GEMM: Y = X @ W^T + b ------------------
// X: n x 256, W: 256 x 256 (row = output channel), Y: n x 256.
// grid = (n/16, 4), block = 128 (4 waves); wave w handles col tile 4*by+w.
__global__ __launch_bounds__(128)
void gemm_qkv_wmma(const float* __restrict__ X, const float* __restrict__ W,
                   const float* __restrict__ bias, float* __restrict__ Y) {
  const int wave = threadIdx.x >> 5;
  const int lane = threadIdx.x & 31;
  const int m0   = blockIdx.x * 16;
  const int n0   = (blockIdx.y * 4 + wave) * 16;
  const int half = lane >> 4;   // which K-pair this lane holds
  const int lr   = lane & 15;   // row for A, col for B

  const float* xptr = X + (size_t)(m0 + lr) * 256 + half * 2;
  const float* wptr = W + (size_t)(n0 + lr) * 256 + half * 2;

  v8f acc = {0.f, 0.f, 0.f, 0.f, 0.f, 0.f, 0.f, 0.f};
#pragma unroll 8
  for (int k = 0; k < 256; k += 4) {
    v2f a = *(const v2f*)(xptr + k);   // A[m][k..k+1] (this lane's K pair)
    v2f b = *(const v2f*)(wptr + k);   // W[n][k..k+1] == B^T row-major
    acc = __builtin_amdgcn_wmma_f32_16x16x4_f32(
        /*neg_a=*/false, a, /*neg_b=*/false, b,
        /*c_mod=*/(short)0, acc, /*reuse_a=*/false, /*reuse_b=*/false);
  }
  const float bv = bias[n0 + lr];
#pragma unroll
  for (int v = 0; v < 8; ++v) {
    int m = m0 + v + half * 8;                  // C/D layout: VGPR v -> row
    Y[(size_t)m * 256 + n0 + lr] = acc[v] + bv;
  }
}

// ------------------------------- kNN (top-16) ------------------------------
// d = |pi|^2 + |pj|^2 - 2 pi.pj ; keep 16 smallest (stable in index on ties).
__global__ __launch_bounds__(256)
void knn_kernel(const float* __restrict__ p, int* __restrict__ idx, int n) {
  const int i = blockIdx.x * 256 + threadIdx.x;
  const float pix = p[i * 3 + 0], piy = p[i * 3 + 1], piz = p[i * 3 + 2];
  const float sqi = pix * pix + piy * piy + piz * piz;

  float bd[16]; int bi[16];
#pragma unroll
  for (int s = 0; s < 16; ++s) { bd[s] = FLT_MAX; bi[s] = INT_MAX; }

  __shared__ float sp[256][3];
  __shared__ float ssq[256];
  for (int tile = 0; tile < n; tile += 256) {
    __syncthreads();
    {
      const int t = threadIdx.x;
      float a = p[(tile + t) * 3 + 0];
      float b = p[(tile + t) * 3 + 1];
      float c = p[(tile + t) * 3 + 2];
      sp[t][0] = a; sp[t][1] = b; sp[t][2] = c;
      ssq[t] = a * a + b * b + c * c;
    }
    __syncthreads();
    for (int u = 0; u < 256; ++u) {
      float dot = pix * sp[u][0] + piy * sp[u][1] + piz * sp[u][2];
      float d = sqi + ssq[u] - 2.0f * dot;
      if (d < bd[15]) {                    // strict < keeps earlier index on tie
        bd[15] = d; bi[15] = tile + u;
#pragma unroll
        for (int s = 15; s > 0; --s) {     // bubble up (stable: strict <)
          if (bd[s] < bd[s - 1]) {
            float td = bd[s]; bd[s] = bd[s - 1]; bd[s - 1] = td;
            int   ti = bi[s]; bi[s] = bi[s - 1]; bi[s - 1] = ti;
          }
        }
      }
    }
  }
#pragma unroll
  for (int s = 0; s < 16; ++s) idx[(size_t)i * 16 + s] = bi[s];
}

// ------------------------- small helpers -----------------------------------
template <int NSJ>
__device__ __forceinline__ void softmax_inplace(float* w) {
  float m = -FLT_MAX;
#pragma unroll
  for (int j = 0; j < NSJ; ++j) m = fmaxf(m, w[j]);
  float s = 0.f;
#pragma unroll
  for (int j = 0; j < NSJ; ++j) { w[j] = expf(w[j] - m); s += w[j]; }
  const float inv = 1.0f / s;
#pragma unroll
  for (int j = 0; j < NSJ; ++j) w[j] *= inv;
}

__global__ void zero_kernel(float* buf, int count) {
  int t = blockIdx.x * blockDim.x + threadIdx.x;
  if (t < count) buf[t] = 0.f;
}

// mean/var -> affine (A = rstd*g, B = b - mean*A)
__global__ void bn_finalize_kernel(const float* stats, const float* g,
                                   const float* b, float* AB, int C, float invN) {
  int c = blockIdx.x * blockDim.x + threadIdx.x;
  if (c >= C) return;
  float mean = stats[c] * invN;
  float var  = stats[C + c] * invN - mean * mean;
  float rstd = rsqrtf(var + 1e-5f);
  float A = rstd * g[c];
  AB[c] = A;
  AB[C + c] = b[c] - mean * A;
}

// ------------- BN stats for pr2 @ pw1^T + pb1 (3 channels) -----------------
__global__ __launch_bounds__(256)
void pbn_stats_kernel(const float* __restrict__ p, const int* __restrict__ idx,
                      const float* __restrict__ pw1, const float* __restrict__ pb1,
                      float* __restrict__ stats) {
  __shared__ float acc[6];
  if (threadIdx.x < 6) acc[threadIdx.x] = 0.f;
  __syncthreads();
  const int t = blockIdx.x * 256 + threadIdx.x;
  const int i = t >> 3, j = t & 7;
  const int jj = idx[(size_t)i * 16 + j];
  const float pr0 = p[jj * 3 + 0] - p[i * 3 + 0];
  const float pr1 = p[jj * 3 + 1] - p[i * 3 + 1];
  const float pr2 = p[jj * 3 + 2] - p[i * 3 + 2];
#pragma unroll
  for (int d = 0; d < 3; ++d) {
    float u = pr0 * pw1[d * 3 + 0] + pr1 * pw1[d * 3 + 1] + pr2 * pw1[d * 3 + 2] + pb1[d];
    atomicAdd(&acc[d], u);
    atomicAdd(&acc[3 + d], u * u);
  }
  __syncthreads();
  if (threadIdx.x < 6) atomicAdd(&stats[threadIdx.x], acc[threadIdx.x]);
}

// t = relu(bn(pr2 @ pw1^T + pb1))  -> materialize (n*8*3)
__global__ __launch_bounds__(256)
void t_kernel(const float* __restrict__ p, const int* __restrict__ idx,
              const float* __restrict__ pw1, const float* __restrict__ pb1,
              const float* __restrict__ AB, float* __restrict__ tbuf) {
  const int t = blockIdx.x * 256 + threadIdx.x;
  const int i = t >> 3, j = t & 7;
  const int jj = idx[(size_t)i * 16 + j];
  const float pr0 = p[jj * 3 + 0] - p[i * 3 + 0];
  const float pr1 = p[jj * 3 + 1] - p[i * 3 + 1];
  const float pr2 = p[jj * 3 + 2] - p[i * 3 + 2];
#pragma unroll
  for (int d = 0; d < 3; ++d) {
    float u = pr0 * pw1[d * 3 + 0] + pr1 * pw1[d * 3 + 1] + pr2 * pw1[d * 3 + 2] + pb1[d];
    tbuf[(size_t)t * 3 + d] = fmaxf(u * AB[d] + AB[3 + d], 0.f);
  }
}

// ------- shared: load per-neighbor 3-vector (pr or t) + indices ------------
template <int NSJ, bool USET>
__device__ __forceinline__ void load_nbr(const int* idx, const float* p,
                                         const float* tbuf, int i, int c,
                                         float (*pr)[3], int* jidx) {
  if (c < NSJ) {
    int jj = idx[(size_t)i * 16 + c];
    jidx[c] = jj;
    if (USET) {
      pr[c][0] = tbuf[((size_t)i * 8 + c) * 3 + 0];
      pr[c][1] = tbuf[((size_t)i * 8 + c) * 3 + 1];
      pr[c][2] = tbuf[((size_t)i * 8 + c) * 3 + 2];
    } else {
      pr[c][0] = p[jj * 3 + 0] - p[i * 3 + 0];
      pr[c][1] = p[jj * 3 + 1] - p[i * 3 + 1];
      pr[c][2] = p[jj * 3 + 2] - p[i * 3 + 2];
    }
  }
}

// ----- BN1 stats over w0 = xk[idx] - xq + pe  (256 channels) ---------------
template <int NSJ, bool USET>
__global__ __launch_bounds__(256)
void bn1_stats_kernel(const float* __restrict__ xk, const float* __restrict__ xq,
                      const int* __restrict__ idx, const float* __restrict__ p,
                      const float* __restrict__ tbuf,
                      const float* __restrict__ pw2, const float* __restrict__ pb2,
                      float* __restrict__ stats) {
  const int i = blockIdx.x, c = threadIdx.x;
  __shared__ float pr[NSJ][3];
  __shared__ int jidx[NSJ];
  load_nbr<NSJ, USET>(idx, p, tbuf, i, c, pr, jidx);
  __syncthreads();
  const float xqv = xq[(size_t)i * 256 + c];
  const float pwa = pw2[c * 3 + 0], pwb = pw2[c * 3 + 1], pwc = pw2[c * 3 + 2];
  const float pbc = pb2[c];
  float s = 0.f, sq = 0.f;
#pragma unroll
  for (int j = 0; j < NSJ; ++j) {
    float pe = pr[j][0] * pwa + pr[j][1] * pwb + pr[j][2] * pwc + pbc;
    float w0 = xk[(size_t)jidx[j] * 256 + c] - xqv + pe;
    s += w0; sq += w0 * w0;
  }
  atomicAdd(&stats[c], s);
  atomicAdd(&stats[256 + c], sq);
}

// ----- pass2: bn1 -> sm -> relu -> sm -> WMMA @lw1^T -> y + bn2 stats ------
// The 256->32 projection runs on the matrix pipe: waves 0/1 each compute one
// 16x16 N-tile with K=256 via V_WMMA_F32_16X16X4_F32, A from LDS, B from lw1.
template <int NSJ, bool USET>
__global__ __launch_bounds__(256)
void pass2_kernel(const float* __restrict__ xk, const float* __restrict__ xq,
                  const int* __restrict__ idx, const float* __restrict__ p,
                  const float* __restrict__ tbuf,
                  const float* __restrict__ pw2, const float* __restrict__ pb2,
                  const float* __restrict__ bn1AB,
                  const float* __restrict__ lw1, const float* __restrict__ lb1,
                  float* __restrict__ ybuf, float* __restrict__ stats2) {
  const int i = blockIdx.x, c = threadIdx.x;
  __shared__ float pr[NSJ][3];
  __shared__ int jidx[NSJ];
  __shared__ float wl[16][258];   // stride 258: 8B-aligned K-pairs, 2-bank lane stride
  __shared__ float yl[16][32];
  load_nbr<NSJ, USET>(idx, p, tbuf, i, c, pr, jidx);
  __syncthreads();

  const float A = bn1AB[c], B = bn1AB[256 + c];
  const float xqv = xq[(size_t)i * 256 + c];
  const float pwa = pw2[c * 3 + 0], pwb = pw2[c * 3 + 1], pwc = pw2[c * 3 + 2];
  const float pbc = pb2[c];

  float w[NSJ];
#pragma unroll
  for (int j = 0; j < NSJ; ++j) {
    float pe = pr[j][0] * pwa + pr[j][1] * pwb + pr[j][2] * pwc + pbc;
    w[j] = (xk[(size_t)jidx[j] * 256 + c] - xqv + pe) * A + B;
  }
  softmax_inplace<NSJ>(w);
#pragma unroll
  for (int j = 0; j < NSJ; ++j) w[j] = fmaxf(w[j], 0.f);   // relu (faithful)
  softmax_inplace<NSJ>(w);
#pragma unroll
  for (int j = 0; j < NSJ; ++j) wl[j][c] = w[j];
  if (NSJ < 16) {                                          // zero pad rows
#pragma unroll
    for (int j = NSJ; j < 16; ++j) wl[j][c] = 0.f;
  }
  __syncthreads();

  const int wave = c >> 5;
  if (wave < 2) {                 // wave-uniform branch: EXEC all-1s for WMMA
    const int lane = c & 31;
    const int half = lane >> 4;
    const int lr   = lane & 15;
    const int cc   = wave * 16 + lr;                 // output channel (B col)
    const float* lrow = lw1 + (size_t)cc * 256 + half * 2;
    const float* arow = &wl[lr][half * 2];           // A row lr, lane's K pair
    v8f acc = {0.f, 0.f, 0.f, 0.f, 0.f, 0.f, 0.f, 0.f};
#pragma unroll 8
    for (int k = 0; k < 256; k += 4) {
      v2f a = *(const v2f*)(arow + k);               // LDS, 8B aligned
      v2f b = *(const v2f*)(lrow + k);
      acc = __builtin_amdgcn_wmma_f32_16x16x4_f32(
          false, a, false, b, (short)0, acc, false, false);
    }
    const float lbv = lb1[cc];
#pragma unroll
    for (int v = 0; v < 8; ++v) {
      int m = v + half * 8;                          // j row
      if (m < NSJ) {
        float val = acc[v] + lbv;
        ybuf[((size_t)i * NSJ + m) * 32 + cc] = val;
        yl[m][cc] = val;
      }
    }
  }
  __syncthreads();
  if (c < 32) {
    float s = 0.f, sq = 0.f;
#pragma unroll
    for (int j = 0; j < NSJ; ++j) { float v = yl[j][c]; s += v; sq += v * v; }
    atomicAdd(&stats2[c], s);
    atomicAdd(&stats2[32 + c], sq);
  }
}

// ----- pass3: bn2 -> sm -> relu -> sm -> @lw2^T -> sm -> weighted sum ------
template <int NSJ, bool USET>
__global__ __launch_bounds__(256)
void pass3_kernel(const float* __restrict__ xv, const int* __restrict__ idx,
                  const float* __restrict__ p, const float* __restrict__ tbuf,
                  const float* __restrict__ pw2, const float* __restrict__ pb2,
                  const float* __restrict__ ybuf, const float* __restrict__ bn2AB,
                  const float* __restrict__ lw2, const float* __restrict__ lb2,
                  float* __restrict__ out, int outOff) {
  const int i = blockIdx.x, c = threadIdx.x;
  __shared__ float pr[NSJ][3];
  __shared__ int jidx[NSJ];
  __shared__ float wl[NSJ][33];
  __shared__ float w2[NSJ][33];
  load_nbr<NSJ, USET>(idx, p, tbuf, i, c, pr, jidx);

  if (c < 32) {
    float z[NSJ];
    const float A = bn2AB[c], B = bn2AB[32 + c];
#pragma unroll
    for (int j = 0; j < NSJ; ++j) z[j] = ybuf[((size_t)i * NSJ + j) * 32 + c] * A + B;
    softmax_inplace<NSJ>(z);
#pragma unroll
    for (int j = 0; j < NSJ; ++j) z[j] = fmaxf(z[j], 0.f);
    softmax_inplace<NSJ>(z);
#pragma unroll
    for (int j = 0; j < NSJ; ++j) wl[j][c] = z[j];
  }
  __syncthreads();

  for (int o = c; o < NSJ * 32; o += 256) {
    const int j = o >> 5, cc = o & 31;
    const float* lrow = lw2 + cc * 32;
    float acc = lb2[cc];
#pragma unroll
    for (int k = 0; k < 32; ++k) acc += wl[j][k] * lrow[k];
    w2[j][cc] = acc;
  }
  __syncthreads();
  if (c < 32) {
    float z[NSJ];
#pragma unroll
    for (int j = 0; j < NSJ; ++j) z[j] = w2[j][c];
    softmax_inplace<NSJ>(z);
#pragma unroll
    for (int j = 0; j < NSJ; ++j) w2[j][c] = z[j];
  }
  __syncthreads();

  const int c2 = c & 31;                      // channel within group of 32
  const float pwa = pw2[c * 3 + 0], pwb = pw2[c * 3 + 1], pwc = pw2[c * 3 + 2];
  const float pbc = pb2[c];
  float acc = 0.f;
#pragma unroll
  for (int j = 0; j < NSJ; ++j) {
    float pe  = pr[j][0] * pwa + pr[j][1] * pwb + pr[j][2] * pwc + pbc;
    float val = xv[(size_t)jidx[j] * 256 + c] + pe;
    acc += val * w2[j][c2];
  }
  out[(size_t)i * 512 + outOff + c] = acc;
}

// ---------------------------------------------------------------------------
extern "C" void kernel_launch(void* const* d_in, const int* in_sizes, int n_in,
                              void* d_out, int out_size, void* d_ws, size_t ws_size,
                              hipStream_t stream) {
  const float* pp   = (const float*)d_in[0];
  const float* x    = (const float*)d_in[1];
  // d_in[2] 'o' (int64 length) unused
  const float* wq   = (const float*)d_in[3];
  const float* bq   = (const float*)d_in[4];
  const float* wk   = (const float*)d_in[5];
  const float* bk   = (const float*)d_in[6];
  const float* wv   = (const float*)d_in[7];
  const float* bvv  = (const float*)d_in[8];
  const float* pw1  = (const float*)d_in[9];
  const float* pb1  = (const float*)d_in[10];
  const float* pg   = (const float*)d_in[11];
  const float* pbt  = (const float*)d_in[12];
  const float* pw2  = (const float*)d_in[13];
  const float* pb2  = (const float*)d_in[14];
  const float* wg1  = (const float*)d_in[15];
  const float* wbt1 = (const float*)d_in[16];
  const float* lw1  = (const float*)d_in[17];
  const float* lb1  = (const float*)d_in[18];
  const float* wg2  = (const float*)d_in[19];
  const float* wbt2 = (const float*)d_in[20];
  const float* lw2  = (const float*)d_in[21];
  const float* lb2  = (const float*)d_in[22];
  float* out = (float*)d_out;

  const int n = in_sizes[0] / 3;              // 8192
  const size_t nf = (size_t)n * 256;

  // ---- workspace layout (floats) ----
  float* ws   = (float*)d_ws;
  float* xq   = ws;                           // n*256
  float* xk   = xq + nf;                      // n*256
  float* xv   = xk + nf;                      // n*256
  int*   idx  = (int*)(xv + nf);              // n*16 int32
  float* y1   = (float*)(idx + (size_t)n * 16);   // n*16*32
  float* y2   = y1 + (size_t)n * 16 * 32;         // n*8*32
  float* tbuf = y2 + (size_t)n * 8 * 32;          // n*8*3
  float* st   = tbuf + (size_t)n * 8 * 3;
  float* st_pbn = st;            float* ab_pbn = st_pbn + 6;
  float* st_b1a = ab_pbn + 6;    float* ab_b1a = st_b1a + 512;
  float* st_b2a = ab_b1a + 512;  float* ab_b2a = st_b2a + 64;
  float* st_b1b = ab_b2a + 64;   float* ab_b1b = st_b1b + 512;
  float* st_b2b = ab_b1b + 512;  float* ab_b2b = st_b2b + 64;
  const int statsCount = 6 + 6 + 2 * (512 + 512 + 64 + 64);   // zero all

  zero_kernel<<<(statsCount + 255) / 256, 256, 0, stream>>>(st, statsCount);

  // kNN top-16
  knn_kernel<<<n / 256, 256, 0, stream>>>(pp, idx, n);

  // QKV GEMMs (WMMA f32 16x16x4)
  dim3 ggrid(n / 16, 4);
  gemm_qkv_wmma<<<ggrid, 128, 0, stream>>>(x, wq, bq, xq);
  gemm_qkv_wmma<<<ggrid, 128, 0, stream>>>(x, wk, bk, xk);
  gemm_qkv_wmma<<<ggrid, 128, 0, stream>>>(x, wv, bvv, xv);

  // t = relu(bn(pr2 @ pw1^T + pb1))
  pbn_stats_kernel<<<(n * 8) / 256, 256, 0, stream>>>(pp, idx, pw1, pb1, st_pbn);
  bn_finalize_kernel<<<1, 64, 0, stream>>>(st_pbn, pg, pbt, ab_pbn, 3, 1.0f / (n * 8));
  t_kernel<<<(n * 8) / 256, 256, 0, stream>>>(pp, idx, pw1, pb1, ab_pbn, tbuf);

  // ---- branch 1 (ns = 16, pe from raw pr) ----
  bn1_stats_kernel<16, false><<<n, 256, 0, stream>>>(xk, xq, idx, pp, nullptr, pw2, pb2, st_b1a);
  bn_finalize_kernel<<<1, 256, 0, stream>>>(st_b1a, wg1, wbt1, ab_b1a, 256, 1.0f / (n * 16));
  pass2_kernel<16, false><<<n, 256, 0, stream>>>(xk, xq, idx, pp, nullptr, pw2, pb2,
                                                 ab_b1a, lw1, lb1, y1, st_b2a);
  bn_finalize_kernel<<<1, 32, 0, stream>>>(st_b2a, wg2, wbt2, ab_b2a, 32, 1.0f / (n * 16));
  pass3_kernel<16, false><<<n, 256, 0, stream>>>(xv, idx, pp, nullptr, pw2, pb2,
                                                 y1, ab_b2a, lw2, lb2, out, 0);

  // ---- branch 2 (ns = 8, pe from t) ----
  bn1_stats_kernel<8, true><<<n, 256, 0, stream>>>(xk, xq, idx, pp, tbuf, pw2, pb2, st_b1b);
  bn_finalize_kernel<<<1, 256, 0, stream>>>(st_b1b, wg1, wbt1, ab_b1b, 256, 1.0f / (n * 8));
  pass2_kernel<8, true><<<n, 256, 0, stream>>>(xk, xq, idx, pp, tbuf, pw2, pb2,
                                               ab_b1b, lw1, lb1, y2, st_b2b);
  bn_finalize_kernel<<<1, 32, 0, stream>>>(st_b2b, wg2, wbt2, ab_b2b, 32, 1.0f / (n * 8));
  pass3_kernel<8, true><<<n, 256, 0, stream>>>(xv, idx, pp, tbuf, pw2, pb2,
                                               y2, ab_b2b, lw2, lb2, out, 256);
}